// ChannelSELayer_36876589204141
// MI455X (gfx1250) — compile-verified
//
#include <hip/hip_runtime.h>
#include <hip/hip_bf16.h>
#include <math.h>

typedef __attribute__((ext_vector_type(2))) float f32x2;
typedef __attribute__((ext_vector_type(4))) float f32x4;
typedef __attribute__((ext_vector_type(8))) float f32x8;

#define S_SPATIAL (64 * 64 * 64)   // 262144 elements per (b, c) channel
#define NCHUNK    8
#define NBC       256              // B*C = 4*64

// ---------------------------------------------------------------------------
// Kernel 1: partial spatial means. grid=(256, 8), block=256.
// Each block reduces 32768 contiguous elements (8192 f32x4) of one channel
// chunk with non-temporal 128-bit loads (don't pollute L2 with a 256MB stream)
// and writes one partial sum.
// ---------------------------------------------------------------------------
__global__ void se_reduce_partial(const float* __restrict__ x,
                                  float* __restrict__ partials) {
    const int bc    = blockIdx.x;   // 0..255  (b*64 + c)
    const int chunk = blockIdx.y;   // 0..7
    const int tid   = threadIdx.x;  // 0..255

    const f32x4* src = (const f32x4*)(x + (size_t)bc * S_SPATIAL);
    const int base = chunk * 8192 + tid;

    float acc = 0.0f;
#pragma unroll 8
    for (int j = 0; j < 32; ++j) {
        f32x4 v = __builtin_nontemporal_load(src + base + j * 256);
        acc += v.x + v.y + v.z + v.w;
    }

    __shared__ float red[256];
    red[tid] = acc;
    __syncthreads();
#pragma unroll
    for (int off = 128; off > 0; off >>= 1) {
        if (tid < off) red[tid] += red[tid + off];
        __syncthreads();
    }
    if (tid == 0) partials[bc * NCHUNK + chunk] = red[0];
}

// ---------------------------------------------------------------------------
// Kernel 2: single wave32. Finish means -> padded 16x64 A in LDS -> fp32 WMMA
// MLP (Linear+LeakyReLU, Linear+Sigmoid) -> per-batch top-2 channel indices.
// V_WMMA_F32_16X16X4_F32 layouts (ISA 7.12.2):
//   A 16x4:  lanes 0-15 (M=lane): VGPR0=K0, VGPR1=K1; lanes 16-31: K2,K3
//   B 4x16:  lanes 0-15 (N=lane): VGPR0=K0, VGPR1=K1; lanes 16-31: K2,K3
//   C/D 16x16: VGPR r -> M=r (lanes 0-15, N=lane) / M=r+8 (lanes 16-31)
// ---------------------------------------------------------------------------
__global__ void se_mlp_topk(const float* __restrict__ partials,
                            const float* __restrict__ W1,
                            const float* __restrict__ b1,
                            const float* __restrict__ W2,
                            const float* __restrict__ b2,
                            int* __restrict__ idx_out) {
    __shared__ float Abuf[16 * 64];   // A matrix (batch rows 0..3, rest zero)
    __shared__ float Hbuf[16 * 64];   // hidden activations

    const int lane = threadIdx.x;     // 0..31, exactly one wave

    // Zero-pad A (rows 4..15 must be zero).
    for (int i = lane; i < 16 * 64; i += 32) Abuf[i] = 0.0f;
    __syncthreads();

    // Finish the mean: 8 (b,c) entries per lane, 8 partials each.
#pragma unroll
    for (int j = 0; j < 8; ++j) {
        const int bc = lane * 8 + j;
        float s = 0.0f;
#pragma unroll
        for (int k = 0; k < NCHUNK; ++k) s += partials[bc * NCHUNK + k];
        const int b = bc >> 6, c = bc & 63;
        Abuf[b * 64 + c] = s * (1.0f / (float)S_SPATIAL);
    }
    __syncthreads();

    const int m     = lane & 15;               // A-row / B-col within tile
    const int khalf = (lane >> 4) << 1;        // 0 (lanes 0-15) or 2 (16-31)
    const int rbase = (lane >> 4) << 3;        // C/D row offset 0 or 8

    // ---- Layer 1: H = leaky_relu(A @ W1^T + b1) ----
    for (int nt = 0; nt < 4; ++nt) {
        const int n = nt * 16 + m;             // output channel
        f32x8 acc = {};
#pragma unroll
        for (int k0 = 0; k0 < 64; k0 += 4) {
            const int kk = k0 + khalf;
            f32x2 a;  a.x  = Abuf[m * 64 + kk];  a.y  = Abuf[m * 64 + kk + 1];
            // B[k][n] = W1[n][k]  (h = y @ W1.T)
            f32x2 bb; bb.x = W1[n * 64 + kk];    bb.y = W1[n * 64 + kk + 1];
            acc = __builtin_amdgcn_wmma_f32_16x16x4_f32(
                false, a, false, bb, (short)0, acc, false, false);
        }
        const float bias = b1[n];
#pragma unroll
        for (int r = 0; r < 8; ++r) {
            float v = acc[r] + bias;
            v = (v > 0.0f) ? v : 0.01f * v;     // LeakyReLU(0.01)
            Hbuf[(r + rbase) * 64 + n] = v;
        }
    }
    __syncthreads();

    // ---- Layer 2: S = sigmoid(H @ W2^T + b2), written back into Abuf ----
    for (int nt = 0; nt < 4; ++nt) {
        const int n = nt * 16 + m;
        f32x8 acc = {};
#pragma unroll
        for (int k0 = 0; k0 < 64; k0 += 4) {
            const int kk = k0 + khalf;
            f32x2 a;  a.x  = Hbuf[m * 64 + kk];  a.y  = Hbuf[m * 64 + kk + 1];
            f32x2 bb; bb.x = W2[n * 64 + kk];    bb.y = W2[n * 64 + kk + 1];
            acc = __builtin_amdgcn_wmma_f32_16x16x4_f32(
                false, a, false, bb, (short)0, acc, false, false);
        }
        const float bias = b2[n];
#pragma unroll
        for (int r = 0; r < 8; ++r) {
            const float v = acc[r] + bias;
            Abuf[(r + rbase) * 64 + n] = 1.0f / (1.0f + __expf(-v));
        }
    }
    __syncthreads();

    // ---- Top-2 per batch row (first-occurrence tie-break, as lax.top_k) ----
    if (lane < 4) {
        const float* srow = &Abuf[lane * 64];
        int i1 = 0; float v1 = srow[0];
        for (int c = 1; c < 64; ++c)
            if (srow[c] > v1) { v1 = srow[c]; i1 = c; }
        int i2 = -1; float v2 = -3.4e38f;
        for (int c = 0; c < 64; ++c) {
            if (c == i1) continue;
            if (srow[c] > v2) { v2 = srow[c]; i2 = c; }
        }
        idx_out[lane * 2 + 0] = i1;
        idx_out[lane * 2 + 1] = i2;
    }
}

// ---------------------------------------------------------------------------
// Kernel 3: gather the selected channels. grid=(64, 8), block=256.
// Each y-block copies 1/64 of one selected channel (f32x4 NT copy).
// ---------------------------------------------------------------------------
__global__ void se_gather(const float* __restrict__ x,
                          const int* __restrict__ idx,
                          float* __restrict__ out) {
    const int br = blockIdx.y;        // b*2 + r, 0..7
    const int b  = br >> 1;
    const int c  = idx[br];

    const f32x4* src = (const f32x4*)(x + ((size_t)(b * 64 + c)) * S_SPATIAL);
    f32x4*       dst = (f32x4*)(out + (size_t)br * S_SPATIAL);

    const int base = blockIdx.x * 1024 + threadIdx.x;
#pragma unroll
    for (int j = 0; j < 4; ++j) {
        f32x4 v = __builtin_nontemporal_load(src + base + j * 256);
        __builtin_nontemporal_store(v, dst + base + j * 256);
    }
}

// ---------------------------------------------------------------------------
extern "C" void kernel_launch(void* const* d_in, const int* in_sizes, int n_in,
                              void* d_out, int out_size, void* d_ws, size_t ws_size,
                              hipStream_t stream) {
    const float* x  = (const float*)d_in[0];
    const float* W1 = (const float*)d_in[1];
    const float* b1 = (const float*)d_in[2];
    const float* W2 = (const float*)d_in[3];
    const float* b2 = (const float*)d_in[4];
    float* out = (float*)d_out;

    float* partials = (float*)d_ws;                                  // 256*8 f32
    int*   idx      = (int*)((char*)d_ws + NBC * NCHUNK * sizeof(float)); // 8 ints

    se_reduce_partial<<<dim3(NBC, NCHUNK), 256, 0, stream>>>(x, partials);
    se_mlp_topk<<<1, 32, 0, stream>>>(partials, W1, b1, W2, b2, idx);
    se_gather<<<dim3(64, 8), 256, 0, stream>>>(x, idx, out);
}